// XDOG_68212670595460
// MI455X (gfx1250) — compile-verified
//
#include <hip/hip_runtime.h>
#include <math.h>

#ifndef __has_builtin
#define __has_builtin(x) 0
#endif

#if __has_builtin(__builtin_amdgcn_global_load_async_to_lds_b128)
#define USE_ASYNC_LDS 1
#else
#define USE_ASYNC_LDS 0
#endif

typedef float v2f __attribute__((ext_vector_type(2)));
typedef float v8f __attribute__((ext_vector_type(8)));
typedef int   v4i __attribute__((ext_vector_type(4)));

#if USE_ASYNC_LDS
typedef v4i __attribute__((address_space(1)))* gv4i_ptr;   // global b128 payload
typedef v4i __attribute__((address_space(3)))* lv4i_ptr;   // LDS b128 payload
#endif

#define IMG_W   1024
#define IMG_H   1024
#define N_IMG   32
#define TILE    64
#define PAD     12
#define XROWS   96          // 88 valid halo rows, padded to 6x16 for WMMA row-chunks
#define XCOLS   88
#define XPITCH  92          // mult. of 4 (b128-aligned), 92%64=28 -> conflict-free cols
#define HPITCH  101         // 101%64=37, inverse 45: no 2-element-offset collisions
#define GAMMA   0.95f
#define THRESH  -0.5f
#define PHI     1.0e9f

__global__ __launch_bounds__(256) void xdog_wmma_kernel(
    const float* __restrict__ x,
    const float* __restrict__ w1,
    const float* __restrict__ w2,
    float* __restrict__ out)
{
    // LDS: input tile, transposed horizontal results (2 filters), padded 1D filters
    __shared__ __align__(16) float Xs[XROWS * XPITCH];   // 35,328 B
    __shared__ __align__(16) float HT[2][TILE * HPITCH]; // 51,712 B (col-major: [col][row])
    __shared__ float U[2][64];                           // 1D filters at +16, zero-padded

    const int tid    = threadIdx.x;
    const int lane   = tid & 31;
    const int lanelo = lane & 15;       // N / M index within half
    const int laneh  = lane >> 4;       // half-wave select
    const int wave   = tid >> 5;        // 0..7

    const int x0 = blockIdx.x * TILE;
    const int y0 = blockIdx.y * TILE;
    const size_t img = (size_t)blockIdx.z * (size_t)(IMG_W * IMG_H);

    // ---- recover separable 1D factors: u[j] = W2D[12][j] / sqrt(W2D[12][12]) ----
    if (tid < 64) {
        float v1 = 0.0f, v2 = 0.0f;
        if (tid >= 16 && tid < 16 + 25) {
            int j = tid - 16;
            v1 = w1[12 * 25 + j] * (1.0f / sqrtf(w1[12 * 25 + 12]));
            v2 = w2[12 * 25 + j] * (1.0f / sqrtf(w2[12 * 25 + 12]));
        }
        U[0][tid] = v1;
        U[1][tid] = v2;
    }

    // ---- load 88x88 halo tile into LDS; rows 88..95 zeroed ----
    const bool interior = (x0 >= PAD) && (x0 + TILE + PAD <= IMG_W) &&
                          (y0 >= PAD) && (y0 + TILE + PAD <= IMG_H);
#if USE_ASYNC_LDS
    if (interior) {
        // Fast path: direct global->LDS async DMA (no VGPR staging), b128 chunks.
        const float* gbase = x + img + (size_t)(y0 - PAD) * IMG_W + (x0 - PAD);
        for (int t = tid; t < 88 * 22; t += 256) {
            int r  = t / 22;
            int c4 = (t % 22) * 4;
            const float* src = gbase + (size_t)r * IMG_W + c4;
            float*       dst = &Xs[r * XPITCH + c4];
            __builtin_amdgcn_global_load_async_to_lds_b128(
                (gv4i_ptr)(void*)src, (lv4i_ptr)(void*)dst, 0, 0);
        }
        for (int t = tid; t < 8 * XCOLS; t += 256)
            Xs[(88 + t / XCOLS) * XPITCH + (t % XCOLS)] = 0.0f;
#if __has_builtin(__builtin_amdgcn_s_wait_asynccnt)
        __builtin_amdgcn_s_wait_asynccnt(0);
#else
        asm volatile("s_wait_asynccnt 0x0" ::: "memory");
#endif
    } else
#endif
    {
        // Border (or fallback) path: zero-padded scalar loads.
        for (int idx = tid; idx < XROWS * XCOLS; idx += 256) {
            int r = idx / XCOLS, c = idx % XCOLS;
            int gy = y0 - PAD + r;
            int gx = x0 - PAD + c;
            float v = 0.0f;
            if (r < 88 && gy >= 0 && gy < IMG_H && gx >= 0 && gx < IMG_W)
                v = x[img + (size_t)gy * IMG_W + gx];
            Xs[r * XPITCH + c] = v;
        }
    }
    __syncthreads();

    // =====================================================================
    // Horizontal pass:  H[y, n] = sum_m Xs[y, 16c+m] * T[m, n],  T[m,n]=u[m-n]
    // D(16x16) = A(16x40 input rows) x B(40x16 Toeplitz), 10 x wmma f32 16x16x4
    // 48 tasks: (rchunk 0..5) x (cchunk 0..3) x (filter 0..1), 6 per wave
    // =====================================================================
    for (int i = 0; i < 6; ++i) {
        const int task = wave + 8 * i;          // 0..47
        const int f = task & 1;
        const int c = (task >> 1) & 3;
        const int r = task >> 3;

        v8f acc = {0.0f, 0.0f, 0.0f, 0.0f, 0.0f, 0.0f, 0.0f, 0.0f};
        const float* arow = &Xs[(16 * r + lanelo) * XPITCH + 16 * c + 2 * laneh];
        const float* tp   = &U[f][16 + 2 * laneh - lanelo];
#pragma unroll
        for (int s = 0; s < 10; ++s) {
            v2f A = { arow[4 * s], arow[4 * s + 1] };           // input patch (M=lanelo rows)
            v2f B = { tp[4 * s],  tp[4 * s + 1] };              // Toeplitz weights (N=lanelo)
            acc = __builtin_amdgcn_wmma_f32_16x16x4_f32(
                false, A, false, B, (short)0, acc, false, false);
        }
        // D element (i=v+8*laneh, n=lanelo) -> HT[f][col=16c+n][row=16r+i]
        float* hp = &HT[f][(16 * c + lanelo) * HPITCH + 16 * r + 8 * laneh];
#pragma unroll
        for (int v = 0; v < 8; ++v) hp[v] = acc[v];
    }
    __syncthreads();

    // =====================================================================
    // Vertical pass:  out[i, n] = sum_m T[i, m] * H[16t+m, 16xc+n],  T[i,m]=u[m-i]
    // D(16x16) = A(16x40 Toeplitz) x B(40x16 H columns), both filters per task
    // 16 position tasks: (t 0..3) x (xc 0..3), 2 per wave
    // =====================================================================
    for (int i = 0; i < 2; ++i) {
        const int task = wave + 8 * i;          // 0..15
        const int t  = task >> 2;
        const int xc = task & 3;

        v8f a1 = {0.0f, 0.0f, 0.0f, 0.0f, 0.0f, 0.0f, 0.0f, 0.0f};
        v8f a2 = {0.0f, 0.0f, 0.0f, 0.0f, 0.0f, 0.0f, 0.0f, 0.0f};
        const float* tp0 = &U[0][16 + 2 * laneh - lanelo];
        const float* tp1 = &U[1][16 + 2 * laneh - lanelo];
        const float* h0  = &HT[0][(16 * xc + lanelo) * HPITCH + 16 * t + 2 * laneh];
        const float* h1  = &HT[1][(16 * xc + lanelo) * HPITCH + 16 * t + 2 * laneh];
#pragma unroll
        for (int s = 0; s < 10; ++s) {
            v2f A0 = { tp0[4 * s], tp0[4 * s + 1] };            // Toeplitz (M=lanelo out-rows)
            v2f B0 = { h0[4 * s],  h0[4 * s + 1] };             // H columns (N=lanelo)
            a1 = __builtin_amdgcn_wmma_f32_16x16x4_f32(
                false, A0, false, B0, (short)0, a1, false, false);
            v2f A1 = { tp1[4 * s], tp1[4 * s + 1] };
            v2f B1 = { h1[4 * s],  h1[4 * s + 1] };
            a2 = __builtin_amdgcn_wmma_f32_16x16x4_f32(
                false, A1, false, B1, (short)0, a2, false, false);
        }

        // Fused DoG + threshold + 1 + tanh(phi*d); D element (v+8*laneh, lanelo)
        const int gx = x0 + 16 * xc + lanelo;
#pragma unroll
        for (int v = 0; v < 8; ++v) {
            float d = a1[v] - GAMMA * a2[v];
            d = (d > THRESH) ? d : 1.0f;
            float o = 1.0f + tanhf(PHI * d);
            int gy = y0 + 16 * t + 8 * laneh + v;
            out[img + (size_t)gy * IMG_W + gx] = o;
        }
    }
}

extern "C" void kernel_launch(void* const* d_in, const int* in_sizes, int n_in,
                              void* d_out, int out_size, void* d_ws, size_t ws_size,
                              hipStream_t stream) {
    const float* x  = (const float*)d_in[0];
    const float* w1 = (const float*)d_in[1];
    const float* w2 = (const float*)d_in[2];
    float* out = (float*)d_out;

    dim3 grid(IMG_W / TILE, IMG_H / TILE, N_IMG);   // 16 x 16 x 32
    dim3 block(256);
    xdog_wmma_kernel<<<grid, block, 0, stream>>>(x, w1, w2, out);
}